// MultivariateTSModel_79525614452906
// MI455X (gfx1250) — compile-verified
//
#include <hip/hip_runtime.h>
#include <hip/hip_bf16.h>
#include <math.h>

#define B_ 32
#define T_ 256
#define C_ 64
#define EMB_ 32
#define HID_ 64
#define G3_ 192
#define CC_ 4096   // C*C
#define H1DIM 2048
#define H2DIM 1024

typedef float v2f __attribute__((ext_vector_type(2)));
typedef float v8f __attribute__((ext_vector_type(8)));

__device__ __forceinline__ float sigmoidf_(float v) { return 1.0f / (1.0f + __expf(-v)); }

// ---- f32 WMMA 16x16x4 tile helpers (layouts per CDNA5 ISA 7.12.2) ----
// A operand (16x4): lane m = lane%16 holds row, VGPR j holds K = j + 2*(lane/16)
__device__ __forceinline__ v2f wmma_ld_a(const float* src, int stride, int row0, int k0, int lane) {
  int m = lane & 15, hl = lane >> 4;
  const float* p = src + (row0 + m) * stride + k0 + 2 * hl;
  v2f a; a.x = p[0]; a.y = p[1];
  return a;
}
// B operand (4x16) where B[k][n] = W[n][k], W row-major [N][K] (i.e. X @ W^T)
__device__ __forceinline__ v2f wmma_ld_bt(const float* W, int stride, int n0, int k0, int lane) {
  int m = lane & 15, hl = lane >> 4;
  const float* p = W + (n0 + m) * stride + k0 + 2 * hl;
  v2f b; b.x = p[0]; b.y = p[1];
  return b;
}
// B operand (4x16) from row-major B[k][n] directly
__device__ __forceinline__ v2f wmma_ld_b(const float* Bm, int stride, int k0, int n0, int lane) {
  int m = lane & 15, hl = lane >> 4;
  v2f b;
  b.x = Bm[(k0 + 2 * hl) * stride + n0 + m];
  b.y = Bm[(k0 + 2 * hl + 1) * stride + n0 + m];
  return b;
}
// D store (16x16 f32): VGPR i -> row = i + 8*(lane/16), col = lane%16
__device__ __forceinline__ void wmma_st(float* dst, int stride, int row0, int col0, int lane, v8f c) {
  int m = lane & 15, hl = lane >> 4;
#pragma unroll
  for (int i = 0; i < 8; i++) dst[(row0 + i + 8 * hl) * stride + col0 + m] = c[i];
}
__device__ __forceinline__ v8f wmma4(v2f a, v2f b, v8f c) {
  return __builtin_amdgcn_wmma_f32_16x16x4_f32(false, a, false, b, (short)0, c, false, false);
}
__device__ __forceinline__ v8f splat8(float v) {
  v8f c;
#pragma unroll
  for (int i = 0; i < 8; i++) c[i] = v;
  return c;
}

// ---- gfx1250 async global->LDS DMA (ASYNCcnt-tracked, no VGPR staging) ----
// Flat LDS addresses carry the LDS byte offset in bits [31:0] (ISA 10.2 aperture rules).
__device__ __forceinline__ void async_ld16(float* lds_ptr, const float* gptr) {
  uint32_t lds_addr = (uint32_t)(uintptr_t)lds_ptr;
  asm volatile("global_load_async_to_lds_b128 %0, %1, off"
               :
               : "v"(lds_addr), "v"(gptr)
               : "memory");
}
__device__ __forceinline__ void async_wait0() {
  asm volatile("s_wait_asynccnt 0x0" ::: "memory");
}

// ---------------- prep: alpha/beta for rank-1 input projection ----------------
__global__ void prep_kernel(const float* __restrict__ emb_w, const float* __restrict__ emb_b,
                            const float* __restrict__ w_ih, const float* __restrict__ b_ih,
                            float* __restrict__ alpha, float* __restrict__ beta,
                            float* __restrict__ recloss) {
  int n = threadIdx.x;
  if (n < G3_) {
    float a = 0.f, bt = 0.f;
#pragma unroll
    for (int e = 0; e < EMB_; e++) {
      float w = w_ih[n * EMB_ + e];
      a += emb_w[e] * w;
      bt += emb_b[e] * w;
    }
    alpha[n] = a;
    beta[n] = bt + b_ih[n];
  }
  if (n == 0) *recloss = 0.0f;
}

// ---------------- phase 1: GRU + attention-mixing recurrence ----------------
// one block per batch; 8 waves; LDS = 4 x (64x64) f32 = 64 KB
__global__ __launch_bounds__(256) void gru_mix_kernel(
    const float* __restrict__ x, const float* __restrict__ w_hh, const float* __restrict__ b_hh,
    const float* __restrict__ q_w, const float* __restrict__ q_b, const float* __restrict__ k_w,
    const float* __restrict__ k_b, const float* __restrict__ gate_b,
    const float* __restrict__ pred_w, const float* __restrict__ pred_b,
    const float* __restrict__ alpha, const float* __restrict__ beta,
    float* __restrict__ out_dnc, float* __restrict__ out_mix, float* __restrict__ out_recloss) {
  __shared__ float sm[4 * 64 * 64];
  float* g0 = sm;                 // gh_r -> q -> reduce scratch
  float* g1 = sm + 4096;          // gh_z -> k
  float* g2 = sm + 8192;          // gh_n -> transfer
  float* hB = sm + 12288;         // h / h_new (persistent)

  const int b = blockIdx.x;
  const int tid = threadIdx.x;
  const int lane = tid & 31;
  const int wave = tid >> 5;

  float dnc_acc[16];
#pragma unroll
  for (int i = 0; i < 16; i++) dnc_acc[i] = 0.f;
  float rec_acc = 0.f;

  // h0 = 0
#pragma unroll
  for (int i = 0; i < 16; i++) hB[tid + 256 * i] = 0.f;
  __syncthreads();

  for (int t = 0; t < T_; t++) {
    // ---- stage A: gh = h @ w_hh^T + b_hh  -> [64 x 192] in g0|g1|g2 ----
    for (int s = 0; s < 6; s++) {
      int tt = wave * 6 + s;          // 48 tiles
      int mt = tt & 3, nt = tt >> 2;  // nt in 0..11
      int n0g = nt * 16;
      v8f c = splat8(b_hh[n0g + (lane & 15)]);
      for (int kk = 0; kk < HID_; kk += 4) {
        v2f a = wmma_ld_a(hB, 64, mt * 16, kk, lane);
        v2f bb = wmma_ld_bt(w_hh, HID_, n0g, kk, lane);
        c = wmma4(a, bb, c);
      }
      wmma_st(sm + (nt >> 2) * 4096, 64, mt * 16, (nt & 3) * 16, lane, c);
    }
    __syncthreads();

    // ---- stage B: GRU elementwise; h_new overwrites hB ----
#pragma unroll
    for (int i = 0; i < 16; i++) {
      int e = tid + 256 * i;
      int cr = e >> 6, j = e & 63;
      float xv = x[(size_t)b * T_ * C_ + (size_t)t * C_ + cr];
      float ir = xv * alpha[j] + beta[j];
      float iz = xv * alpha[64 + j] + beta[64 + j];
      float in_ = xv * alpha[128 + j] + beta[128 + j];
      float r = sigmoidf_(ir + g0[e]);
      float z = sigmoidf_(iz + g1[e]);
      float ng = tanhf(in_ + r * g2[e]);
      hB[e] = (1.0f - z) * ng + z * hB[e];
    }
    __syncthreads();

    // ---- stage C: q = h_new@q_w^T + q_b -> g0 ; k = h_new@k_w^T + k_b -> g1 ----
    for (int s = 0; s < 4; s++) {
      int jj = wave * 4 + s;              // 32 tile-jobs
      int which = jj >> 4;                // 0 = q, 1 = k
      int tile = jj & 15;
      int mt = tile & 3, nt = tile >> 2;
      const float* W = which ? k_w : q_w;
      const float* bv = which ? k_b : q_b;
      v8f c = splat8(bv[nt * 16 + (lane & 15)]);
      for (int kk = 0; kk < HID_; kk += 4) {
        v2f a = wmma_ld_a(hB, 64, mt * 16, kk, lane);
        v2f bb = wmma_ld_bt(W, HID_, nt * 16, kk, lane);
        c = wmma4(a, bb, c);
      }
      wmma_st(which ? g1 : g0, 64, mt * 16, nt * 16, lane, c);
    }
    __syncthreads();

    // ---- stage D: transfer = q @ k^T -> g2 ----
    for (int s = 0; s < 2; s++) {
      int tile = wave * 2 + s;
      int mt = tile & 3, nt = tile >> 2;
      v8f c = splat8(0.0f);
      for (int kk = 0; kk < HID_; kk += 4) {
        v2f a = wmma_ld_a(g0, 64, mt * 16, kk, lane);
        v2f bb = wmma_ld_bt(g1, 64, nt * 16, kk, lane);  // B[k][n] = kmat[n][k]
        c = wmma4(a, bb, c);
      }
      wmma_st(g2, 64, mt * 16, nt * 16, lane, c);
    }
    __syncthreads();

    // ---- stage E: Frobenius norm of transfer (reduce via g0, q is dead) ----
    {
      float p = 0.f;
#pragma unroll
      for (int i = 0; i < 16; i++) {
        float v = g2[tid + 256 * i];
        p += v * v;
      }
      __syncthreads();
      g0[tid] = p;
      __syncthreads();
      for (int s = 128; s > 0; s >>= 1) {
        if (tid < s) g0[tid] += g0[tid + s];
        __syncthreads();
      }
    }
    float inv_norm = 1.0f / sqrtf(g0[0]);

    // ---- stage F: tanh + gate, emit mixing, accumulate DNC ----
#pragma unroll
    for (int i = 0; i < 16; i++) {
      int e = tid + 256 * i;
      float v = tanhf(g2[e] * inv_norm);
      float g = sigmoidf_(fabsf(v) + gate_b[e]);
      v *= g;
      g2[e] = v;
      out_mix[((size_t)(b * T_ + t)) * CC_ + e] = v;
      dnc_acc[i] += v;
    }
    __syncthreads();

    // ---- stage G: h_next = transfer @ h_new (regs, then overwrite hB) ----
    v8f acc[2];
    for (int s = 0; s < 2; s++) {
      int tile = wave * 2 + s;
      int mt = tile & 3, nt = tile >> 2;
      v8f c = splat8(0.0f);
      for (int kk = 0; kk < HID_; kk += 4) {
        v2f a = wmma_ld_a(g2, 64, mt * 16, kk, lane);
        v2f bb = wmma_ld_b(hB, 64, kk, nt * 16, lane);  // B[k][n] = h_new[k][n]
        c = wmma4(a, bb, c);
      }
      acc[s] = c;
    }
    __syncthreads();
    for (int s = 0; s < 2; s++) {
      int tile = wave * 2 + s;
      int mt = tile & 3, nt = tile >> 2;
      wmma_st(hB, 64, mt * 16, nt * 16, lane, acc[s]);
    }
    __syncthreads();

    // ---- stage H: prediction + rec-loss accumulation (hidden_states[:,1:]) ----
    if (t >= 1 && tid < C_) {
      float p = pred_b[0];
#pragma unroll
      for (int j = 0; j < HID_; j++) p += hB[tid * 64 + j] * pred_w[j];
      float d = p - x[(size_t)b * T_ * C_ + (size_t)t * C_ + tid];
      rec_acc += d * d;
    }
  }

  // DNC = mean over t
#pragma unroll
  for (int i = 0; i < 16; i++) out_dnc[(size_t)b * CC_ + tid + 256 * i] = dnc_acc[i] * (1.0f / T_);

  // rec_loss reduction + atomic accumulate
  __syncthreads();
  g0[tid] = rec_acc;
  __syncthreads();
  for (int s = 128; s > 0; s >>= 1) {
    if (tid < s) g0[tid] += g0[tid + s];
    __syncthreads();
  }
  if (tid == 0) atomicAdd(out_recloss, g0[0] * (1.0f / ((float)B_ * (T_ - 1) * C_)));
}

// ---------------- phase 2: D = act(A @ W^T + bias), 64x64 tile per block ----------------
// Double-buffered LDS staging via gfx1250 async global->LDS (ASYNCcnt).
__device__ __forceinline__ void gemm_stage_async(const float* __restrict__ A,
                                                 const float* __restrict__ W, float* sA, float* sW,
                                                 int m0, int n0, int K, int kc, int tid) {
#pragma unroll
  for (int p = 0; p < 2; p++) {
    int f = (tid + p * 256) * 4;       // flat float index into 64x32 chunk
    int r = f >> 5, cl = f & 31;       // row, col (col multiple of 4 -> 16B aligned)
    async_ld16(sA + f, A + (size_t)(m0 + r) * K + kc + cl);
    async_ld16(sW + f, W + (size_t)(n0 + r) * K + kc + cl);
  }
}

__global__ __launch_bounds__(256) void gemm_relu_kernel(
    const float* __restrict__ A, const float* __restrict__ W, const float* __restrict__ bias,
    float* __restrict__ D, int M, int N, int K, int do_relu) {
  __shared__ __align__(16) float smem[2 * 2 * 64 * 32];  // {A,W} x double buffer, 32 KB
  const int tid = threadIdx.x;
  const int lane = tid & 31;
  const int wave = tid >> 5;
  const int m0 = blockIdx.y * 64;
  const int n0 = blockIdx.x * 64;
  const int mt = wave & 3;
  const int ntb = (wave >> 2) * 2;

  v8f acc[2];
#pragma unroll
  for (int s = 0; s < 2; s++) acc[s] = splat8(bias[n0 + (ntb + s) * 16 + (lane & 15)]);

  // prologue: async-stage chunk 0
  gemm_stage_async(A, W, smem, smem + 2048, m0, n0, K, 0, tid);
  async_wait0();
  __syncthreads();

  for (int kc = 0; kc < K; kc += 32) {
    const int cur = (kc >> 5) & 1;
    float* sA = smem + cur * 4096;
    float* sW = smem + cur * 4096 + 2048;

    // overlap: issue async copy of next chunk into the other buffer
    if (kc + 32 < K) {
      int nxt = cur ^ 1;
      gemm_stage_async(A, W, smem + nxt * 4096, smem + nxt * 4096 + 2048, m0, n0, K, kc + 32, tid);
    }

    for (int kk = 0; kk < 32; kk += 4) {
      v2f a = wmma_ld_a(sA, 32, mt * 16, kk, lane);
#pragma unroll
      for (int s = 0; s < 2; s++) {
        v2f bb = wmma_ld_bt(sW, 32, (ntb + s) * 16, kk, lane);
        acc[s] = wmma4(a, bb, acc[s]);
      }
    }

    async_wait0();     // my async copies for buf[next] are in LDS
    __syncthreads();   // everyone done reading buf[cur] and done staging buf[next]
  }

  const int m = lane & 15, hl = lane >> 4;
#pragma unroll
  for (int s = 0; s < 2; s++) {
    int col = n0 + (ntb + s) * 16 + m;
#pragma unroll
    for (int i = 0; i < 8; i++) {
      int row = m0 + mt * 16 + i + 8 * hl;
      float v = acc[s][i];
      if (do_relu) v = fmaxf(v, 0.0f);
      D[(size_t)row * N + col] = v;
    }
  }
}

// ---------------- logits: mean over t of h2 @ clf3_w^T + b3 ----------------
__global__ __launch_bounds__(256) void logits_kernel(const float* __restrict__ h2,
                                                     const float* __restrict__ w3,
                                                     const float* __restrict__ b3,
                                                     float* __restrict__ out) {
  __shared__ float red[512];
  const int b = blockIdx.x, tid = threadIdx.x;
  float a0 = 0.f, a1 = 0.f;
  for (int t = 0; t < T_; t++) {
    const float* row = h2 + ((size_t)(b * T_ + t)) * H2DIM;
    for (int j = tid; j < H2DIM; j += 256) {
      float v = row[j];
      a0 += v * w3[j];
      a1 += v * w3[H2DIM + j];
    }
  }
  red[tid] = a0;
  red[256 + tid] = a1;
  __syncthreads();
  for (int s = 128; s > 0; s >>= 1) {
    if (tid < s) {
      red[tid] += red[tid + s];
      red[256 + tid] += red[256 + tid + s];
    }
    __syncthreads();
  }
  if (tid == 0) {
    out[b * 2 + 0] = red[0] * (1.0f / T_) + b3[0];
    out[b * 2 + 1] = red[256] * (1.0f / T_) + b3[1];
  }
}

extern "C" void kernel_launch(void* const* d_in, const int* in_sizes, int n_in, void* d_out,
                              int out_size, void* d_ws, size_t ws_size, hipStream_t stream) {
  (void)in_sizes; (void)n_in; (void)out_size; (void)ws_size;
  const float* x      = (const float*)d_in[0];
  const float* emb_w  = (const float*)d_in[1];
  const float* emb_b  = (const float*)d_in[2];
  const float* w_ih   = (const float*)d_in[3];
  const float* w_hh   = (const float*)d_in[4];
  const float* b_ih   = (const float*)d_in[5];
  const float* b_hh   = (const float*)d_in[6];
  const float* q_w    = (const float*)d_in[7];
  const float* q_b    = (const float*)d_in[8];
  const float* k_w    = (const float*)d_in[9];
  const float* k_b    = (const float*)d_in[10];
  const float* gate_b = (const float*)d_in[11];
  const float* pred_w = (const float*)d_in[12];
  const float* pred_b = (const float*)d_in[13];
  const float* clf1_w = (const float*)d_in[14];
  const float* clf1_b = (const float*)d_in[15];
  const float* clf2_w = (const float*)d_in[16];
  const float* clf2_b = (const float*)d_in[17];
  const float* clf3_w = (const float*)d_in[18];
  const float* clf3_b = (const float*)d_in[19];

  float* out     = (float*)d_out;
  float* logits  = out;                                   // [B,2] = 64
  float* dnc     = out + 64;                              // [B,64,64] = 131072
  float* mix     = out + 64 + (size_t)B_ * CC_;           // [B,T,64,64] = 33554432
  float* recloss = out + 64 + (size_t)B_ * CC_ + (size_t)B_ * T_ * CC_;

  float* ws    = (float*)d_ws;
  float* alpha = ws;         // 192
  float* beta  = ws + 192;   // 192
  float* h1    = ws + 512;                                  // [8192, 2048]
  float* h2    = h1 + (size_t)(B_ * T_) * H1DIM;            // [8192, 1024]

  hipLaunchKernelGGL(prep_kernel, dim3(1), dim3(192), 0, stream,
                     emb_w, emb_b, w_ih, b_ih, alpha, beta, recloss);

  hipLaunchKernelGGL(gru_mix_kernel, dim3(B_), dim3(256), 0, stream,
                     x, w_hh, b_hh, q_w, q_b, k_w, k_b, gate_b, pred_w, pred_b,
                     alpha, beta, dnc, mix, recloss);

  hipLaunchKernelGGL(gemm_relu_kernel, dim3(H1DIM / 64, (B_ * T_) / 64), dim3(256), 0, stream,
                     mix, clf1_w, clf1_b, h1, B_ * T_, H1DIM, CC_, 1);

  hipLaunchKernelGGL(gemm_relu_kernel, dim3(H2DIM / 64, (B_ * T_) / 64), dim3(256), 0, stream,
                     h1, clf2_w, clf2_b, h2, B_ * T_, H2DIM, H1DIM, 1);

  hipLaunchKernelGGL(logits_kernel, dim3(B_), dim3(256), 0, stream, h2, clf3_w, clf3_b, logits);
}